// EMMA_31078383354699
// MI455X (gfx1250) — compile-verified
//
#include <hip/hip_runtime.h>
#include <hip/hip_bf16.h>
#include <math.h>

// Problem constants (match reference)
constexpr int Bn = 256, Sn = 128, Hn = 768, Zn = 128;
constexpr float COS_EPS_F = 1e-8f;
#define FINF __builtin_inff()

typedef __attribute__((ext_vector_type(2))) float v2f;
typedef __attribute__((ext_vector_type(8))) float v8f;

// ---------------------------------------------------------------------------
// WMMA f32 GEMM: C[M,N] = A[M,K] @ B + bias (optional relu)
// TRANSB=0: Bm is row-major K x N.  TRANSB=1: Bm is row-major N x K (use B^T).
// One wave computes a 16x16 tile with V_WMMA_F32_16X16X4_F32 (K-step 4).
// ISA layouts (CDNA5 ISA):
//   A 16x4 : lanes 0-15 -> M=lane, VGPR{0,1}=K{0,1}; lanes 16-31 -> K{2,3}
//   B 4x16 : lanes 0-15 -> N=lane, VGPR{0,1}=K{0,1}; lanes 16-31 -> K{2,3}
//   C 16x16: VGPR r: lanes 0-15 -> M=r, lanes 16-31 -> M=r+8; N = lane%16
// ---------------------------------------------------------------------------
template <int TRANSB, int RELU>
__global__ void __launch_bounds__(128) wmma_gemm_kernel(
    const float* __restrict__ A, const float* __restrict__ Bm,
    const float* __restrict__ bias, float* __restrict__ C,
    int M, int N, int K) {
  const int lane  = threadIdx.x & 31;
  const int wave  = threadIdx.x >> 5;
  const int m0    = blockIdx.x * 16;
  const int n0    = blockIdx.y * 64 + wave * 16;   // wave-uniform
  if (n0 >= N) return;                             // keeps EXEC all-1s per wave
  const int half  = lane >> 4;                     // 0 or 1
  const int idx16 = lane & 15;

  v8f acc = {0.f, 0.f, 0.f, 0.f, 0.f, 0.f, 0.f, 0.f};
  const float* __restrict__ Ap = A + (size_t)(m0 + idx16) * K + 2 * half;

  for (int k0 = 0; k0 < K; k0 += 4) {
    v2f a, bfrag;
    a.x = Ap[k0];
    a.y = Ap[k0 + 1];
    const int kb = k0 + 2 * half;
    if (TRANSB) {
      const float* p = Bm + (size_t)(n0 + idx16) * K + kb;
      bfrag.x = p[0];
      bfrag.y = p[1];
    } else {
      const float* p = Bm + (size_t)kb * N + (n0 + idx16);
      bfrag.x = p[0];
      bfrag.y = p[N];
    }
    acc = __builtin_amdgcn_wmma_f32_16x16x4_f32(
        /*neg_a=*/false, a, /*neg_b=*/false, bfrag,
        /*c_mod=*/(short)0, acc, /*reuse_a=*/false, /*reuse_b=*/false);
  }

  const float bv = bias ? bias[n0 + idx16] : 0.f;
#pragma unroll
  for (int r = 0; r < 8; ++r) {
    float v = acc[r] + bv;
    if (RELU) v = fmaxf(v, 0.f);
    C[(size_t)(m0 + r + 8 * half) * N + (n0 + idx16)] = v;
  }
}

static inline void gemm(hipStream_t st, const float* A, const float* Bm,
                        const float* bias, float* C, int M, int N, int K,
                        bool transB, bool relu) {
  dim3 g(M / 16, (N + 63) / 64), b(128);
  if (transB) {
    if (relu) wmma_gemm_kernel<1, 1><<<g, b, 0, st>>>(A, Bm, bias, C, M, N, K);
    else      wmma_gemm_kernel<1, 0><<<g, b, 0, st>>>(A, Bm, bias, C, M, N, K);
  } else {
    if (relu) wmma_gemm_kernel<0, 1><<<g, b, 0, st>>>(A, Bm, bias, C, M, N, K);
    else      wmma_gemm_kernel<0, 0><<<g, b, 0, st>>>(A, Bm, bias, C, M, N, K);
  }
}

// ---------------------------------------------------------------------------
// Pass 1: single streaming pass over sen/des producing masked-max extracts
// and sequence means. Block per b, coalesced over H.
// ---------------------------------------------------------------------------
__global__ void __launch_bounds__(256) reduce_s_kernel(
    const float* __restrict__ sen, const float* __restrict__ des,
    const int* __restrict__ m1g, const int* __restrict__ m2g,
    float* __restrict__ e1raw, float* __restrict__ e2raw,
    float* __restrict__ enraw, float* __restrict__ senmean,
    float* __restrict__ desmean) {
  __shared__ int m1s[Sn], m2s[Sn];
  __shared__ int firstIdx, lastIdx;
  const int b = blockIdx.x, t = threadIdx.x;
  if (t == 0) { firstIdx = Sn; lastIdx = -1; }
  __syncthreads();
  if (t < Sn) {
    int a = m1g[b * Sn + t], c = m2g[b * Sn + t];
    m1s[t] = a; m2s[t] = c;
    if (a + c > 0) { atomicMin(&firstIdx, t); atomicMax(&lastIdx, t); }
  }
  __syncthreads();
  const int fi = firstIdx, la = lastIdx;

  for (int h = t; h < Hn; h += 256) {
    float mx1 = -FINF, mx2 = -FINF, mxe = -FINF, s1 = 0.f, s2 = 0.f;
    const float* ps = sen + (size_t)b * Sn * Hn + h;
    const float* pd = des + (size_t)b * Sn * Hn + h;
    for (int s = 0; s < Sn; ++s) {
      const float xv = ps[(size_t)s * Hn];
      const float dv = pd[(size_t)s * Hn];
      mx1 = fmaxf(mx1, m1s[s] ? xv : 0.f);
      mx2 = fmaxf(mx2, m2s[s] ? xv : 0.f);
      const bool in = (s >= fi) && (s <= la);
      mxe = fmaxf(mxe, in ? xv : 0.f);
      s1 += xv; s2 += dv;
    }
    e1raw[b * Hn + h] = mx1;
    e2raw[b * Hn + h] = mx2;
    enraw[b * Hn + h] = mxe;
    senmean[b * Hn + h] = s1 * (1.f / Sn);
    desmean[b * Hn + h] = s2 * (1.f / Sn);
  }
}

// ---------------------------------------------------------------------------
// Row-wise concat of up to 3 sources into dst (row-major, rows = gridDim.x)
// ---------------------------------------------------------------------------
__global__ void concat3_kernel(float* __restrict__ dst,
                               const float* __restrict__ s0, int w0,
                               const float* __restrict__ s1, int w1,
                               const float* __restrict__ s2, int w2) {
  const int b = blockIdx.x, t = threadIdx.x, W = w0 + w1 + w2;
  for (int i = t; i < W; i += blockDim.x) {
    float v;
    if (i < w0)            v = s0[(size_t)b * w0 + i];
    else if (i < w0 + w1)  v = s1[(size_t)b * w1 + (i - w0)];
    else                   v = s2[(size_t)b * w2 + (i - w0 - w1)];
    dst[(size_t)b * W + i] = v;
  }
}

// z = mu + eps*exp(0.5*lv); per-block KL partial (deterministic, no atomics)
__global__ void __launch_bounds__(256) zvae_kernel(
    const float* __restrict__ mu, const float* __restrict__ lv,
    const float* __restrict__ eps, float* __restrict__ z,
    float* __restrict__ klp) {
  __shared__ float sm[256];
  const int i = blockIdx.x * 256 + threadIdx.x;   // exact multiple
  const float m = mu[i], l = lv[i];
  z[i] = m + eps[i] * expf(0.5f * l);
  sm[threadIdx.x] = 0.5f * (m * m + expf(l) - l - 1.f);
  __syncthreads();
  for (int o = 128; o > 0; o >>= 1) {
    if (threadIdx.x < o) sm[threadIdx.x] += sm[threadIdx.x + o];
    __syncthreads();
  }
  if (threadIdx.x == 0) klp[blockIdx.x] = sm[0];
}

// Fused attention pooling: scores (relu(des·t_p + bias)*2) -> softmax -> apply
__global__ void __launch_bounds__(128) pool_kernel(
    const float* __restrict__ des, const float* __restrict__ t1,
    const float* __restrict__ t2, const float* __restrict__ t3,
    const float* __restrict__ bias1, const float* __restrict__ bias2,
    const float* __restrict__ bias3, float* __restrict__ p1,
    float* __restrict__ p2, float* __restrict__ p3) {
  __shared__ float q[3][Hn];
  __shared__ float sc[3][Sn];
  __shared__ float aw[3][Sn];
  __shared__ float red[3];
  const int b = blockIdx.x, t = threadIdx.x;
  for (int i = t; i < Hn; i += 128) {
    q[0][i] = t1[b * Hn + i]; q[1][i] = t2[b * Hn + i]; q[2][i] = t3[b * Hn + i];
  }
  __syncthreads();
  {  // phase 1: thread t <-> sequence position s=t
    const float* row = des + ((size_t)b * Sn + t) * Hn;
    float d0 = 0.f, d1 = 0.f, d2 = 0.f;
    for (int i = 0; i < Hn; ++i) {
      const float x = row[i];
      d0 += x * q[0][i]; d1 += x * q[1][i]; d2 += x * q[2][i];
    }
    sc[0][t] = fmaxf(d0 + bias1[t], 0.f) * 2.f;
    sc[1][t] = fmaxf(d1 + bias2[t], 0.f) * 2.f;
    sc[2][t] = fmaxf(d2 + bias3[t], 0.f) * 2.f;
  }
  __syncthreads();
  if (t < 3) {
    float m = -FINF;
    for (int s = 0; s < Sn; ++s) m = fmaxf(m, sc[t][s]);
    float sum = 0.f;
    for (int s = 0; s < Sn; ++s) { float e = expf(sc[t][s] - m); aw[t][s] = e; sum += e; }
    red[t] = sum;
  }
  __syncthreads();
  for (int i = t; i < 3 * Sn; i += 128) aw[i / Sn][i % Sn] /= red[i / Sn];
  __syncthreads();
  for (int j = 0; j < Hn / 128; ++j) {  // phase 2: weighted sum over s
    const int h = t + 128 * j;
    float a0 = 0.f, a1 = 0.f, a2 = 0.f;
    const float* col = des + (size_t)b * Sn * Hn + h;
    for (int s = 0; s < Sn; ++s) {
      const float x = col[(size_t)s * Hn];
      a0 += aw[0][s] * x; a1 += aw[1][s] * x; a2 += aw[2][s] * x;
    }
    p1[b * Hn + h] = a0; p2[b * Hn + h] = a1; p3[b * Hn + h] = a2;
  }
}

// L2-normalize rows in place (denominator max(norm,1e-12))
__global__ void __launch_bounds__(256) l2rows_kernel(float* __restrict__ X, int D) {
  __shared__ float sm[256];
  __shared__ float scale;
  const int b = blockIdx.x, t = threadIdx.x;
  float ss = 0.f;
  for (int i = t; i < D; i += 256) { float v = X[(size_t)b * D + i]; ss += v * v; }
  sm[t] = ss; __syncthreads();
  for (int o = 128; o > 0; o >>= 1) { if (t < o) sm[t] += sm[t + o]; __syncthreads(); }
  if (t == 0) scale = 1.f / fmaxf(sqrtf(sm[0]), 1e-12f);
  __syncthreads();
  for (int i = t; i < D; i += 256) X[(size_t)b * D + i] *= scale;
}

__global__ void __launch_bounds__(256) rownorm_kernel(
    const float* __restrict__ X, int D, float* __restrict__ out) {
  __shared__ float sm[256];
  const int b = blockIdx.x, t = threadIdx.x;
  float ss = 0.f;
  for (int i = t; i < D; i += 256) { float v = X[(size_t)b * D + i]; ss += v * v; }
  sm[t] = ss; __syncthreads();
  for (int o = 128; o > 0; o >>= 1) { if (t < o) sm[t] += sm[t + o]; __syncthreads(); }
  if (t == 0) out[b] = sqrtf(sm[0]);
}

// G[i,j] /= max(na[i]*nb[j], COS_EPS)
__global__ void cos_scale_kernel(float* __restrict__ G,
                                 const float* __restrict__ na,
                                 const float* __restrict__ nb) {
  const int b = blockIdx.x, t = threadIdx.x;
  G[b * Bn + t] /= fmaxf(na[b] * nb[t], COS_EPS_F);
}

// per-row kk-th largest index via iterative argmax (ties -> smaller index)
__global__ void __launch_bounds__(256) topk_kernel(
    const float* __restrict__ G, const int* __restrict__ negc,
    int* __restrict__ outIdx) {
  __shared__ float vals[Bn];
  __shared__ float rv[256];
  __shared__ int ri[256];
  const int b = blockIdx.x, t = threadIdx.x;
  vals[t] = G[b * Bn + t];
  __syncthreads();
  int kk = 3 + negc[0];
  if (kk < 1) kk = 1;
  if (kk > 64) kk = 64;
  int last = -1;
  for (int it = 0; it < kk; ++it) {
    rv[t] = vals[t]; ri[t] = t; __syncthreads();
    for (int o = 128; o > 0; o >>= 1) {
      if (t < o) {
        if (rv[t + o] > rv[t] || (rv[t + o] == rv[t] && ri[t + o] < ri[t])) {
          rv[t] = rv[t + o]; ri[t] = ri[t + o];
        }
      }
      __syncthreads();
    }
    last = ri[0];
    __syncthreads();
    if (t == 0) vals[last] = -FINF;
    __syncthreads();
  }
  if (t == 0) outIdx[b] = last;
}

// neg_extra[b] = cos(sen_vec[b], [desmean[sec[b]], sen_ent[esec[b]]])
__global__ void __launch_bounds__(256) negextra_kernel(
    const float* __restrict__ sen_vec, const float* __restrict__ desmean,
    const float* __restrict__ sen_ent, const int* __restrict__ sec,
    const int* __restrict__ esec, const float* __restrict__ n_svec,
    float* __restrict__ negx) {
  __shared__ float sd[256], sn[256];
  const int b = blockIdx.x, t = threadIdx.x;
  const int sb = sec[b], eb = esec[b];
  const float* sv = sen_vec + (size_t)b * (4 * Hn);
  float dot = 0.f, nsq = 0.f;
  for (int i = t; i < 4 * Hn; i += 256) {
    const float dv = (i < Hn) ? desmean[(size_t)sb * Hn + i]
                              : sen_ent[(size_t)eb * (3 * Hn) + (i - Hn)];
    dot += sv[i] * dv;
    nsq += dv * dv;
  }
  sd[t] = dot; sn[t] = nsq; __syncthreads();
  for (int o = 128; o > 0; o >>= 1) {
    if (t < o) { sd[t] += sd[t + o]; sn[t] += sn[t + o]; }
    __syncthreads();
  }
  if (t == 0)
    negx[b] = sd[0] / fmaxf(n_svec[b] * sqrtf(sn[0]), COS_EPS_F);
}

// ori_p[b] = LSE({cos[b,:]/0.02} U {negx[b]/0.02}) - cos[b,b]/0.02
__global__ void __launch_bounds__(256) ori_kernel(
    const float* __restrict__ cosfull, const float* __restrict__ negx,
    float* __restrict__ orip) {
  __shared__ float sm[256];
  const int b = blockIdx.x, t = threadIdx.x;
  const float v = cosfull[b * Bn + t] * 50.f;
  const float extra = negx[b] * 50.f;
  float lm = (t == 0) ? fmaxf(v, extra) : v;
  sm[t] = lm; __syncthreads();
  for (int o = 128; o > 0; o >>= 1) { if (t < o) sm[t] = fmaxf(sm[t], sm[t + o]); __syncthreads(); }
  const float m = sm[0]; __syncthreads();
  float e = expf(v - m);
  if (t == 0) e += expf(extra - m);
  sm[t] = e; __syncthreads();
  for (int o = 128; o > 0; o >>= 1) { if (t < o) sm[t] += sm[t + o]; __syncthreads(); }
  if (t == 0) {
    const float lse = m + logf(sm[0]);
    orip[b] = lse - cosfull[b * Bn + b] * 50.f;
  }
}

__global__ void __launch_bounds__(256) huber_kernel(
    const float* __restrict__ recon, const float* __restrict__ desmean,
    float* __restrict__ hbp) {
  __shared__ float sm[256];
  const int i = blockIdx.x * 256 + threadIdx.x;  // exact multiple
  const float e = recon[i] - desmean[i];
  const float ae = fabsf(e);
  sm[threadIdx.x] = (ae < 1.f) ? 0.5f * e * e : (ae - 0.5f);
  __syncthreads();
  for (int o = 128; o > 0; o >>= 1) {
    if (threadIdx.x < o) sm[threadIdx.x] += sm[threadIdx.x + o];
    __syncthreads();
  }
  if (threadIdx.x == 0) hbp[blockIdx.x] = sm[0];
}

__global__ void __launch_bounds__(256) finalize_kernel(
    const float* __restrict__ orip, const float* __restrict__ klp,
    const float* __restrict__ hbp, float* __restrict__ out) {
  __shared__ float sm[256];
  const int t = threadIdx.x;
  float so = orip[t];
  float sk = (t < 128) ? klp[t] : 0.f;
  float sh = 0.f;
  for (int i = t; i < 768; i += 256) sh += hbp[i];
  sm[t] = so; __syncthreads();
  for (int o = 128; o > 0; o >>= 1) { if (t < o) sm[t] += sm[t + o]; __syncthreads(); }
  const float ori = sm[0]; __syncthreads();
  sm[t] = sk; __syncthreads();
  for (int o = 128; o > 0; o >>= 1) { if (t < o) sm[t] += sm[t + o]; __syncthreads(); }
  const float kl = sm[0]; __syncthreads();
  sm[t] = sh; __syncthreads();
  for (int o = 128; o > 0; o >>= 1) { if (t < o) sm[t] += sm[t + o]; __syncthreads(); }
  if (t == 0) {
    const float hb = sm[0];
    out[0] = ori / (float)Bn + 0.1f * (hb / (float)(Bn * Hn)) + 0.05f * (kl / (float)Bn);
  }
}

// ---------------------------------------------------------------------------
extern "C" void kernel_launch(void* const* d_in, const int* in_sizes, int n_in,
                              void* d_out, int out_size, void* d_ws, size_t ws_size,
                              hipStream_t stream) {
  const float* sen = (const float*)d_in[0];
  const float* des = (const float*)d_in[1];
  const float* eps = (const float*)d_in[2];
  const int*   m1  = (const int*)d_in[3];
  const int*   m2  = (const int*)d_in[4];
  const float* sen_e1_W = (const float*)d_in[5];  const float* sen_e1_b = (const float*)d_in[6];
  const float* sen_e2_W = (const float*)d_in[7];  const float* sen_e2_b = (const float*)d_in[8];
  const float* sen_en_W = (const float*)d_in[9];  const float* sen_en_b = (const float*)d_in[10];
  const float* des_e1 = (const float*)d_in[11];
  const float* des_e2 = (const float*)d_in[12];
  const float* des_en = (const float*)d_in[13];
  const float* enc1_W = (const float*)d_in[14];   const float* enc1_b = (const float*)d_in[15];
  const float* mu_W = (const float*)d_in[16];     const float* mu_b = (const float*)d_in[17];
  const float* lv_W = (const float*)d_in[18];     const float* lv_b = (const float*)d_in[19];
  const float* dec1_W = (const float*)d_in[20];   const float* dec1_b = (const float*)d_in[21];
  const float* dout_W = (const float*)d_in[22];   const float* dout_b = (const float*)d_in[23];
  const float* bias1 = (const float*)d_in[24];
  const float* bias2 = (const float*)d_in[25];
  const float* bias3 = (const float*)d_in[26];
  const int*   negc  = (const int*)d_in[27];
  float* out = (float*)d_out;

  // --- workspace carve (floats) ---
  float* w = (float*)d_ws;
  size_t o = 0;
  auto take = [&](size_t n) { float* p = w + o; o += n; return p; };
  const size_t BH = (size_t)Bn * Hn;
  float* e1raw   = take(BH);
  float* e2raw   = take(BH);
  float* enraw   = take(BH);
  float* senmean = take(BH);
  float* desmean = take(BH);
  float* e1vec   = take(BH);
  float* e2vec   = take(BH);
  float* envec   = take(BH);
  float* xcat    = take(3 * BH);
  float* hid     = take(BH);
  float* mu      = take((size_t)Bn * Zn);
  float* lv      = take((size_t)Bn * Zn);
  float* zbuf    = take((size_t)Bn * Zn);
  float* ycat    = take((size_t)Bn * (Zn + 2 * Hn));
  float* h2      = take((size_t)Bn * 512);
  float* recon   = take(BH);
  float* t1      = take(BH);
  float* t2      = take(BH);
  float* t3      = take(BH);
  float* p1      = take(BH);
  float* p2      = take(BH);
  float* p3      = take(BH);
  float* sen_ent = take(3 * BH);
  float* des_ent = take(3 * BH);
  float* sen_vec = take(4 * BH);
  float* des_vec = take(4 * BH);
  float* n_smean = take(Bn);
  float* n_dmean = take(Bn);
  float* n_sent  = take(Bn);
  float* n_svec  = take(Bn);
  float* n_dvec  = take(Bn);
  float* Gf      = take((size_t)Bn * Bn);
  float* Ge      = take((size_t)Bn * Bn);
  float* Gfull   = take((size_t)Bn * Bn);
  float* negx    = take(Bn);
  float* orip    = take(Bn);
  float* klp     = take(128);
  float* hbp     = take(768);
  int* sec  = (int*)take(Bn);
  int* esec = (int*)take(Bn);

  // 1) single pass over sen/des: extracts + means
  reduce_s_kernel<<<Bn, 256, 0, stream>>>(sen, des, m1, m2, e1raw, e2raw, enraw,
                                          senmean, desmean);
  // 2) entity projections (WMMA)
  gemm(stream, e1raw, sen_e1_W, sen_e1_b, e1vec, Bn, Hn, Hn, false, false);
  gemm(stream, e2raw, sen_e2_W, sen_e2_b, e2vec, Bn, Hn, Hn, false, false);
  gemm(stream, enraw, sen_en_W, sen_en_b, envec, Bn, Hn, Hn, false, false);
  // 3) encoder
  concat3_kernel<<<Bn, 256, 0, stream>>>(xcat, e1vec, Hn, e2vec, Hn, desmean, Hn);
  gemm(stream, xcat, enc1_W, enc1_b, hid, Bn, Hn, 3 * Hn, false, true);
  gemm(stream, hid, mu_W, mu_b, mu, Bn, Zn, Hn, false, false);
  gemm(stream, hid, lv_W, lv_b, lv, Bn, Zn, Hn, false, false);
  zvae_kernel<<<(Bn * Zn) / 256, 256, 0, stream>>>(mu, lv, eps, zbuf, klp);
  // 4) decoder
  concat3_kernel<<<Bn, 256, 0, stream>>>(ycat, zbuf, Zn, e1vec, Hn, e2vec, Hn);
  gemm(stream, ycat, dec1_W, dec1_b, h2, Bn, 512, Zn + 2 * Hn, false, true);
  gemm(stream, h2, dout_W, dout_b, recon, Bn, Hn, 512, false, false);
  // 5) bilinear pooling queries: t_p[b,i] = sum_j W[i,j] q[b,j]  (q @ W^T)
  gemm(stream, e1vec, des_e1, nullptr, t1, Bn, Hn, Hn, true, false);
  gemm(stream, e2vec, des_e2, nullptr, t2, Bn, Hn, Hn, true, false);
  gemm(stream, envec, des_en, nullptr, t3, Bn, Hn, Hn, true, false);
  pool_kernel<<<Bn, 128, 0, stream>>>(des, t1, t2, t3, bias1, bias2, bias3,
                                      p1, p2, p3);
  // 6) entity vectors (l2-normalized concats) and full vectors
  concat3_kernel<<<Bn, 256, 0, stream>>>(sen_ent, e1vec, Hn, e2vec, Hn, envec, Hn);
  l2rows_kernel<<<Bn, 256, 0, stream>>>(sen_ent, 3 * Hn);
  concat3_kernel<<<Bn, 256, 0, stream>>>(des_ent, p1, Hn, p2, Hn, p3, Hn);
  l2rows_kernel<<<Bn, 256, 0, stream>>>(des_ent, 3 * Hn);
  concat3_kernel<<<Bn, 256, 0, stream>>>(sen_vec, senmean, Hn, sen_ent, 3 * Hn,
                                         senmean, 0);
  concat3_kernel<<<Bn, 256, 0, stream>>>(des_vec, desmean, Hn, des_ent, 3 * Hn,
                                         desmean, 0);
  // 7) norms for cosine denominators
  rownorm_kernel<<<Bn, 256, 0, stream>>>(senmean, Hn, n_smean);
  rownorm_kernel<<<Bn, 256, 0, stream>>>(desmean, Hn, n_dmean);
  rownorm_kernel<<<Bn, 256, 0, stream>>>(sen_ent, 3 * Hn, n_sent);
  rownorm_kernel<<<Bn, 256, 0, stream>>>(sen_vec, 4 * Hn, n_svec);
  rownorm_kernel<<<Bn, 256, 0, stream>>>(des_vec, 4 * Hn, n_dvec);
  // 8) cosine matrices via WMMA Gram (A @ B^T) + scale
  gemm(stream, senmean, desmean, nullptr, Gf, Bn, Bn, Hn, true, false);
  cos_scale_kernel<<<Bn, Bn, 0, stream>>>(Gf, n_smean, n_dmean);
  topk_kernel<<<Bn, 256, 0, stream>>>(Gf, negc, sec);
  gemm(stream, sen_ent, sen_ent, nullptr, Ge, Bn, Bn, 3 * Hn, true, false);
  cos_scale_kernel<<<Bn, Bn, 0, stream>>>(Ge, n_sent, n_sent);
  topk_kernel<<<Bn, 256, 0, stream>>>(Ge, negc, esec);
  gemm(stream, sen_vec, des_vec, nullptr, Gfull, Bn, Bn, 4 * Hn, true, false);
  cos_scale_kernel<<<Bn, Bn, 0, stream>>>(Gfull, n_svec, n_dvec);
  // 9) extra negative + loss terms
  negextra_kernel<<<Bn, 256, 0, stream>>>(sen_vec, desmean, sen_ent, sec, esec,
                                          n_svec, negx);
  ori_kernel<<<Bn, 256, 0, stream>>>(Gfull, negx, orip);
  huber_kernel<<<(Bn * Hn) / 256, 256, 0, stream>>>(recon, desmean, hbp);
  finalize_kernel<<<1, 256, 0, stream>>>(orip, klp, hbp, out);
}